// SoccerPaynet_51685636440661
// MI455X (gfx1250) — compile-verified
//
#include <hip/hip_runtime.h>

// QRE smoothed-fictitious-play on a hard-coded 4-sequence treeplex.
// One row per lane; all state lives in VGPRs for all 1000 iterations.
// Memory: one NT f32 load per row, three NT b128 stores per row.

#define TAU_F   0.1f
#define N_ITERS 1000
#define EPS_F   1e-8f

typedef __attribute__((ext_vector_type(4))) float v4f;  // native clang vector

__device__ __forceinline__ float fast_rcp(float a) {
    return __builtin_amdgcn_rcpf(a);   // v_rcp_f32 (TRANS unit)
}

// Logit (QRE) smoothed best response, one row, registers only.
// Caller applies sign: z = sign * g.
__device__ __forceinline__ void smoothed_br(
    float z0, float z1, float z2, float z3,
    float l0, float l1, float inv_l1,
    float& b0, float& b1, float& b2, float& b3)
{
    // bottom infoset: softmax + logsumexp over {z2, z3}, stabilized like jax.nn
    float t0 = l1 * z2;
    float t1 = l1 * z3;
    float m  = fmaxf(t0, t1);
    float e0 = __expf(t0 - m);          // v_exp_f32
    float e1 = __expf(t1 - m);
    float s  = e0 + e1;
    float rs = fast_rcp(s);
    float p2 = e0 * rs;
    float p3 = e1 * rs;
    float J1 = (m + __logf(s)) * inv_l1;  // logsumexp(t)/l1, v_log_f32

    // top infoset: softmax over {l0*z0, l0*(z1+J1)}
    float a0 = l0 * z0;
    float a1 = l0 * (z1 + J1);
    float mq = fmaxf(a0, a1);
    float f0 = __expf(a0 - mq);
    float f1 = __expf(a1 - mq);
    float rq = fast_rcp(f0 + f1);
    b0 = f0 * rq;
    b1 = f1 * rq;
    b2 = b1 * p2;
    b3 = b1 * p3;
}

__global__ __launch_bounds__(256) void soccer_qre_kernel(
    const float* __restrict__ X,   // [N,1]
    const float* __restrict__ W,   // [4,1]
    const float* __restrict__ P,   // [4,4] row-major
    float* __restrict__ out,       // u[N,4] ++ v[N,4] ++ j[N,4]
    int N)
{
    const int row = blockIdx.x * blockDim.x + threadIdx.x;
    if (row >= N) return;

    // Uniform (wave-invariant) operands -> scalar loads / SGPRs.
    const float w0 = W[0], w1 = W[1], w2 = W[2], w3 = W[3];
    const float P00 = P[0],  P01 = P[1],  P02 = P[2],  P03 = P[3];
    const float P10 = P[4],  P11 = P[5],  P12 = P[6],  P13 = P[7];
    const float P20 = P[8],  P21 = P[9],  P22 = P[10], P23 = P[11];
    const float P30 = P[12], P31 = P[13], P32 = P[14], P33 = P[15];

    // x is read exactly once -> non-temporal load, keep L2 for neighbors.
    const float x = __builtin_nontemporal_load(X + row);

    // j = exp(x * W^T) + EPS ;  Lu = j[0:2], Lv = j[2:4]
    const float j0 = __expf(x * w0) + EPS_F;
    const float j1 = __expf(x * w1) + EPS_F;
    const float j2 = __expf(x * w2) + EPS_F;
    const float j3 = __expf(x * w3) + EPS_F;

    const float Lu0 = j0, Lu1 = j1;
    const float Lv0 = j2, Lv1 = j3;
    // loop-invariant reciprocals: one exact divide each, outside the loop
    const float inv_Lu1 = 1.0f / Lu1;
    const float inv_Lv1 = 1.0f / Lv1;

    float u0 = 0.5f, u1 = 0.5f, u2 = 0.25f, u3 = 0.25f;
    float v0 = 0.5f, v1 = 0.5f, v2 = 0.25f, v3 = 0.25f;

#pragma unroll 2
    for (int it = 0; it < N_ITERS; ++it) {
        // gu = v @ P^T : gu[i] = sum_j P[i][j] * v[j]
        float gu0 = fmaf(P00, v0, fmaf(P01, v1, fmaf(P02, v2, P03 * v3)));
        float gu1 = fmaf(P10, v0, fmaf(P11, v1, fmaf(P12, v2, P13 * v3)));
        float gu2 = fmaf(P20, v0, fmaf(P21, v1, fmaf(P22, v2, P23 * v3)));
        float gu3 = fmaf(P30, v0, fmaf(P31, v1, fmaf(P32, v2, P33 * v3)));
        // gv = u @ P : gv[j] = sum_i u[i] * P[i][j]
        float gv0 = fmaf(P00, u0, fmaf(P10, u1, fmaf(P20, u2, P30 * u3)));
        float gv1 = fmaf(P01, u0, fmaf(P11, u1, fmaf(P21, u2, P31 * u3)));
        float gv2 = fmaf(P02, u0, fmaf(P12, u1, fmaf(P22, u2, P32 * u3)));
        float gv3 = fmaf(P03, u0, fmaf(P13, u1, fmaf(P23, u2, P33 * u3)));

        // bu and bv are fully independent chains -> 2-way ILP for the
        // TRANS unit (v_exp/v_log co-execute while the other chain's VALU runs)
        float bu0, bu1, bu2, bu3;
        float bv0, bv1, bv2, bv3;
        smoothed_br( gu0,  gu1,  gu2,  gu3, Lu0, Lu1, inv_Lu1, bu0, bu1, bu2, bu3);
        smoothed_br(-gv0, -gv1, -gv2, -gv3, Lv0, Lv1, inv_Lv1, bv0, bv1, bv2, bv3);

        u0 = fmaf(TAU_F, bu0 - u0, u0);
        u1 = fmaf(TAU_F, bu1 - u1, u1);
        u2 = fmaf(TAU_F, bu2 - u2, u2);
        u3 = fmaf(TAU_F, bu3 - u3, u3);
        v0 = fmaf(TAU_F, bv0 - v0, v0);
        v1 = fmaf(TAU_F, bv1 - v1, v1);
        v2 = fmaf(TAU_F, bv2 - v2, v2);
        v3 = fmaf(TAU_F, bv3 - v3, v3);
    }

    // Outputs written once, never re-read by us -> NT b128 stores
    // (global_store_b128 th:TH_STORE_NT), keep the 192MB L2 clean.
    v4f uo = { u0, u1, u2, u3 };
    v4f vo = { v0, v1, v2, v3 };
    v4f jo = { j0, j1, j2, j3 };
    v4f* up = reinterpret_cast<v4f*>(out) + row;
    v4f* vp = reinterpret_cast<v4f*>(out + 4ull * (unsigned)N) + row;
    v4f* jp = reinterpret_cast<v4f*>(out + 8ull * (unsigned)N) + row;
    __builtin_nontemporal_store(uo, up);
    __builtin_nontemporal_store(vo, vp);
    __builtin_nontemporal_store(jo, jp);
}

extern "C" void kernel_launch(void* const* d_in, const int* in_sizes, int n_in,
                              void* d_out, int out_size, void* d_ws, size_t ws_size,
                              hipStream_t stream) {
    const float* x = (const float*)d_in[0];   // [N,1] f32
    const float* W = (const float*)d_in[1];   // [4,1] f32
    const float* P = (const float*)d_in[2];   // [4,4] f32
    float* out = (float*)d_out;               // u ++ v ++ j, each [N,4] f32

    const int N = in_sizes[0];                // 8192
    const int block = 256;                    // 8 wave32s
    const int grid  = (N + block - 1) / block;

    hipLaunchKernelGGL(soccer_qre_kernel, dim3(grid), dim3(block), 0, stream,
                       x, W, P, out, N);
}